// Generator_64613488001734
// MI455X (gfx1250) — compile-verified
//
#include <hip/hip_runtime.h>
#include <math.h>

// Problem constants (from the reference)
#define B_  32
#define T_  64
#define TE_ 64
#define V_  32000
#define E_  256
#define H_  512
#define DOM_ 16
#define STY_ 16
#define HD_ 544           // H + DOM + STY
#define TAU_INV 10.0f

typedef __attribute__((ext_vector_type(2))) float v2f;
typedef __attribute__((ext_vector_type(8))) float v8f;

__device__ __forceinline__ float sigmf(float x) { return 1.0f / (1.0f + expf(-x)); }

// ---------------------------------------------------------------------------
// Generic fp32 WMMA GEMM:  C[M,N] = gather(A) @ op(W) + bias
//   A rows: if a_idx != null, row m of A is  A + a_idx[m*a_idx_stride]*lda
//   w_nk==1: W is (N,K) row-major  -> C = A @ W^T   (ldw = K row stride)
//   w_nk==0: W is (K,N) row-major  -> C = A @ W     (ldw = N row stride)
//
// One wave computes a 16 x (16*NB) strip; A fragment loaded once per k-step
// and reused for NB WMMAs. The k-loop is software-pipelined (double-buffered)
// so each V_WMMA_F32_16X16X4_F32 consumes operands loaded one iteration
// earlier, keeping loads in flight instead of s_wait_loadcnt 0 stalls.
// M, N multiples of 16; K multiple of 4 and >= 8; (N/16) divisible by NB.
// ---------------------------------------------------------------------------
template <int NB>
__global__ void gemm_wmma_f32(const float* __restrict__ A, int lda,
                              const int* __restrict__ a_idx, int a_idx_stride,
                              const float* __restrict__ W, int ldw, int w_nk,
                              const float* __restrict__ bias,
                              float* __restrict__ C, int ldc,
                              int M, int N, int K)
{
    const int lane = threadIdx.x;            // 0..31
    const int tiles_m  = M >> 4;
    const int groups_n = (N >> 4) / NB;
    const int group = blockIdx.x * blockDim.y + threadIdx.y;
    if (group >= tiles_m * groups_n) return; // wave-uniform: EXEC stays all-1s
    const int tm  = group % tiles_m;
    const int tn0 = (group / tiles_m) * NB;

    const int half = lane >> 4;              // 0: K+0/1, 1: K+2/3
    const int l16  = lane & 15;
    const int m    = tm * 16 + l16;          // A row handled by this lane

    const float* arow;
    if (a_idx) arow = A + (size_t)a_idx[(size_t)m * a_idx_stride] * lda;
    else       arow = A + (size_t)m * lda;

    v8f acc[NB];
#pragma unroll
    for (int i = 0; i < NB; ++i) acc[i] = (v8f){};

    if (w_nk) {
        const float* wrow[NB];
#pragma unroll
        for (int i = 0; i < NB; ++i)
            wrow[i] = W + (size_t)((tn0 + i) * 16 + l16) * ldw;

        int kb = 2 * half;                       // prologue: load k-chunk 0
        v2f a;  a.x = arow[kb];  a.y = arow[kb + 1];
        v2f b[NB];
#pragma unroll
        for (int i = 0; i < NB; ++i) { b[i].x = wrow[i][kb]; b[i].y = wrow[i][kb + 1]; }

        for (int k = 4; k < K; k += 4) {
            const int kn = k + 2 * half;         // issue next-chunk loads first
            v2f an; an.x = arow[kn]; an.y = arow[kn + 1];
            v2f bn[NB];
#pragma unroll
            for (int i = 0; i < NB; ++i) { bn[i].x = wrow[i][kn]; bn[i].y = wrow[i][kn + 1]; }
#pragma unroll
            for (int i = 0; i < NB; ++i)         // compute on previous chunk
                acc[i] = __builtin_amdgcn_wmma_f32_16x16x4_f32(
                    false, a, false, b[i], (short)0, acc[i], false, false);
            a = an;
#pragma unroll
            for (int i = 0; i < NB; ++i) b[i] = bn[i];
        }
#pragma unroll
        for (int i = 0; i < NB; ++i)             // epilogue
            acc[i] = __builtin_amdgcn_wmma_f32_16x16x4_f32(
                false, a, false, b[i], (short)0, acc[i], false, false);
    } else {
        int kb = 2 * half;
        v2f a;  a.x = arow[kb];  a.y = arow[kb + 1];
        v2f b[NB];
#pragma unroll
        for (int i = 0; i < NB; ++i) {
            const int n = (tn0 + i) * 16 + l16;
            b[i].x = W[(size_t)kb * ldw + n];
            b[i].y = W[(size_t)(kb + 1) * ldw + n];
        }
        for (int k = 4; k < K; k += 4) {
            const int kn = k + 2 * half;
            v2f an; an.x = arow[kn]; an.y = arow[kn + 1];
            v2f bn[NB];
#pragma unroll
            for (int i = 0; i < NB; ++i) {
                const int n = (tn0 + i) * 16 + l16;
                bn[i].x = W[(size_t)kn * ldw + n];
                bn[i].y = W[(size_t)(kn + 1) * ldw + n];
            }
#pragma unroll
            for (int i = 0; i < NB; ++i)
                acc[i] = __builtin_amdgcn_wmma_f32_16x16x4_f32(
                    false, a, false, b[i], (short)0, acc[i], false, false);
            a = an;
#pragma unroll
            for (int i = 0; i < NB; ++i) b[i] = bn[i];
        }
#pragma unroll
        for (int i = 0; i < NB; ++i)
            acc[i] = __builtin_amdgcn_wmma_f32_16x16x4_f32(
                false, a, false, b[i], (short)0, acc[i], false, false);
    }

#pragma unroll
    for (int i = 0; i < NB; ++i) {
        const int col = (tn0 + i) * 16 + l16;
        const float bv = bias ? bias[col] : 0.0f;
#pragma unroll
        for (int j = 0; j < 8; ++j) {
            const int row = tm * 16 + j + 8 * half;  // D layout: VGPR j -> M=j / M=j+8
            C[(size_t)row * ldc + col] = acc[i][j] + bv;
        }
    }
}

// ---------------------------------------------------------------------------
// GRU gate combine:  h' = (1-z)*n + z*h, optional sequence-length mask.
// ---------------------------------------------------------------------------
__global__ void gru_combine(const float* __restrict__ gi, const float* __restrict__ gh,
                            const float* __restrict__ h_in, float* __restrict__ h_out,
                            const int* __restrict__ lens, int t, int Hsz)
{
    const int idx = blockIdx.x * blockDim.x + threadIdx.x;
    if (idx >= B_ * Hsz) return;
    const int b = idx / Hsz;
    const int j = idx - b * Hsz;
    const float* gib = gi + (size_t)b * 3 * Hsz;
    const float* ghb = gh + (size_t)b * 3 * Hsz;
    const float r  = sigmf(gib[j]           + ghb[j]);
    const float z  = sigmf(gib[Hsz + j]     + ghb[Hsz + j]);
    const float nn = tanhf(gib[2 * Hsz + j] + r * ghb[2 * Hsz + j]);
    const float h  = h_in[idx];
    float hn = (1.0f - z) * nn + z * h;
    if (lens) hn = (t < lens[b]) ? hn : h;
    h_out[idx] = hn;
}

// Zero fill
__global__ void zero_f32(float* __restrict__ p, int n)
{
    const int i = blockIdx.x * blockDim.x + threadIdx.x;
    if (i < n) p[i] = 0.0f;
}

// Build origInfo/tranInfo = concat([sty, dom, h_enc], axis=1)
__global__ void build_info(const float* __restrict__ label, const float* __restrict__ tgtDom,
                           const float* __restrict__ sty_w, const float* __restrict__ sty_b,
                           const float* __restrict__ dom_w, const float* __restrict__ dom_b,
                           const float* __restrict__ h_enc,
                           float* __restrict__ orig, float* __restrict__ tran)
{
    const int idx = blockIdx.x * blockDim.x + threadIdx.x;
    if (idx >= B_ * HD_) return;
    const int b = idx / HD_;
    const int j = idx - b * HD_;
    float o, tr;
    if (j < STY_) {
        const float l = label[b];
        o  = l * sty_w[j] + sty_b[j];
        tr = (1.0f - l) * sty_w[j] + sty_b[j];
    } else if (j < STY_ + DOM_) {
        const int jd = j - STY_;
        const float d = tgtDom[b] * dom_w[jd] + dom_b[jd];
        o = d; tr = d;
    } else {
        const float v = h_enc[(size_t)b * H_ + (j - STY_ - DOM_)];
        o = v; tr = v;
    }
    orig[idx] = o;
    tran[idx] = tr;
}

// Pairwise merge of (max, scaled-sum) pairs in shared memory.
__device__ __forceinline__ void lse_tree_reduce(float* redm, float* reds)
{
    for (int s = 128; s > 0; s >>= 1) {
        if ((int)threadIdx.x < s) {
            const float m1 = redm[threadIdx.x],     s1 = reds[threadIdx.x];
            const float m2 = redm[threadIdx.x + s], s2 = reds[threadIdx.x + s];
            const float M  = fmaxf(m1, m2);
            reds[threadIdx.x] = s1 * expf(m1 - M) + s2 * expf(m2 - M);
            redm[threadIdx.x] = M;
        }
        __syncthreads();
    }
}

// Per-row NLL via single-pass online logsumexp:
//   nll[t*B+b] = mask * (logsumexp(row) - row[tar[b,t]])
__global__ void nll_rows(const float* __restrict__ logits, int t0,
                         const int* __restrict__ tar, const int* __restrict__ decLen,
                         float* __restrict__ nll)
{
    __shared__ float redm[256];
    __shared__ float reds[256];
    const int r = blockIdx.x;
    const int t = t0 + (r >> 5);
    const int b = r & 31;
    const float* row = logits + (size_t)r * V_;

    float mx = -3.4e38f, sum = 0.0f;
    for (int v = threadIdx.x; v < V_; v += 256) {
        const float x = row[v];
        if (x > mx) { sum = sum * expf(mx - x) + 1.0f; mx = x; }
        else        { sum += expf(x - mx); }
    }
    redm[threadIdx.x] = mx; reds[threadIdx.x] = sum; __syncthreads();
    lse_tree_reduce(redm, reds);
    if (threadIdx.x == 0) {
        const float lse = logf(reds[0]) + redm[0];
        const int tok = tar[(size_t)b * T_ + t];
        const float val = lse - row[tok];
        nll[t * B_ + b] = (t < decLen[b]) ? val : 0.0f;
    }
}

// Deterministic final reduction: aeLoss = sum(nll) / B
__global__ void ae_reduce(const float* __restrict__ nll, float* __restrict__ out)
{
    __shared__ float red[256];
    float s = 0.0f;
    for (int i = threadIdx.x; i < T_ * B_; i += 256) s += nll[i];
    red[threadIdx.x] = s; __syncthreads();
    for (int st = 128; st > 0; st >>= 1) {
        if (threadIdx.x < st) red[threadIdx.x] += red[threadIdx.x + st];
        __syncthreads();
    }
    if (threadIdx.x == 0) out[0] = red[0] / (float)B_;
}

// Gumbel softmax row, online logsumexp (2 global passes instead of 3):
//   p = softmax((logits + g) / TAU); writes proba (for the re-embedding GEMM)
//   and the tranSoft output slice (B,T,V layout).
__global__ void gumbel_softmax(const float* __restrict__ logits,
                               const float* __restrict__ g,      // gumbel + t*B*V
                               float* __restrict__ proba,
                               float* __restrict__ outTr, int t)
{
    __shared__ float redm[256];
    __shared__ float reds[256];
    const int b = blockIdx.x;
    const float* row = logits + (size_t)b * V_;
    const float* gb  = g + (size_t)b * V_;

    float mx = -3.4e38f, sum = 0.0f;
    for (int v = threadIdx.x; v < V_; v += 256) {
        const float x = (row[v] + gb[v]) * TAU_INV;
        if (x > mx) { sum = sum * expf(mx - x) + 1.0f; mx = x; }
        else        { sum += expf(x - mx); }
    }
    redm[threadIdx.x] = mx; reds[threadIdx.x] = sum; __syncthreads();
    lse_tree_reduce(redm, reds);

    const float MX  = redm[0];
    const float inv = 1.0f / reds[0];

    for (int v = threadIdx.x; v < V_; v += 256) {
        const float p = expf((row[v] + gb[v]) * TAU_INV - MX) * inv;
        proba[(size_t)b * V_ + v] = p;
        outTr[((size_t)b * T_ + t) * V_ + v] = p;
    }
}

// ---------------------------------------------------------------------------

extern "C" void kernel_launch(void* const* d_in, const int* in_sizes, int n_in,
                              void* d_out, int out_size, void* d_ws, size_t ws_size,
                              hipStream_t stream)
{
    (void)in_sizes; (void)n_in; (void)out_size; (void)ws_size;

    const int*   enc      = (const int*)  d_in[0];
    const int*   encLen   = (const int*)  d_in[1];
    const int*   dec      = (const int*)  d_in[2];
    const int*   decLen   = (const int*)  d_in[3];
    const int*   tar      = (const int*)  d_in[4];
    const float* label    = (const float*)d_in[5];
    const float* tgtDom   = (const float*)d_in[6];
    const float* gumbel   = (const float*)d_in[7];
    const float* emb_w    = (const float*)d_in[8];
    const float* w_ih_enc = (const float*)d_in[9];
    const float* w_hh_enc = (const float*)d_in[10];
    const float* b_ih_enc = (const float*)d_in[11];
    const float* b_hh_enc = (const float*)d_in[12];
    const float* w_ih_dec = (const float*)d_in[13];
    const float* w_hh_dec = (const float*)d_in[14];
    const float* b_ih_dec = (const float*)d_in[15];
    const float* b_hh_dec = (const float*)d_in[16];
    const float* proj_w   = (const float*)d_in[17];
    const float* proj_b   = (const float*)d_in[18];
    const float* dom_w    = (const float*)d_in[19];
    const float* dom_b    = (const float*)d_in[20];
    const float* sty_w    = (const float*)d_in[21];
    const float* sty_b    = (const float*)d_in[22];

    float* outAe = (float*)d_out;
    float* outTr = outAe + 1;

    // Workspace layout (floats); total ~42 MB.
    float* ws    = (float*)d_ws;
    float* f_h   = ws;                                  // 32*512     encoder hidden
    float* f_or  = f_h   + (size_t)B_ * H_;             // 32*544     origInfo
    float* f_tr  = f_or  + (size_t)B_ * HD_;            // 32*544     tranInfo
    float* f_gi  = f_tr  + (size_t)B_ * HD_;            // 32*1632    input gates
    float* f_gh  = f_gi  + (size_t)B_ * 3 * HD_;        // 32*1632    hidden gates
    float* f_hs  = f_gh  + (size_t)B_ * 3 * HD_;        // 64*32*544  TF decoder states
    float* f_nll = f_hs  + (size_t)T_ * B_ * HD_;       // 2048       per-(t,b) nll
    float* f_lg  = f_nll + (size_t)T_ * B_;             // 256*32000  logits scratch
    float* f_hg  = f_lg  + (size_t)8 * B_ * V_;         // 32*544     gumbel hidden
    float* f_x   = f_hg  + (size_t)B_ * HD_;            // 32*256     gumbel input
    float* f_pr  = f_x   + (size_t)B_ * E_;             // 32*32000   proba

    // Pick the widest N-blocking that divides the tile count, keeping
    // control flow wave-uniform (EXEC must stay all-1s around WMMA).
    auto gemm = [&](const float* A, int lda, const int* aidx, int aistride,
                    const float* W, int ldw, int wnk, const float* bias,
                    float* C, int ldc, int M, int N, int K) {
        const int tiles_n = N / 16;
        const int tiles_m = M / 16;
        dim3 blk(32, 4, 1);
        if (tiles_n % 4 == 0) {
            const int groups = tiles_m * (tiles_n / 4);
            gemm_wmma_f32<4><<<dim3((groups + 3) / 4), blk, 0, stream>>>(
                A, lda, aidx, aistride, W, ldw, wnk, bias, C, ldc, M, N, K);
        } else if (tiles_n % 2 == 0) {
            const int groups = tiles_m * (tiles_n / 2);
            gemm_wmma_f32<2><<<dim3((groups + 3) / 4), blk, 0, stream>>>(
                A, lda, aidx, aistride, W, ldw, wnk, bias, C, ldc, M, N, K);
        } else {
            const int groups = tiles_m * tiles_n;
            gemm_wmma_f32<1><<<dim3((groups + 3) / 4), blk, 0, stream>>>(
                A, lda, aidx, aistride, W, ldw, wnk, bias, C, ldc, M, N, K);
        }
    };

    // ---------------- Encoder GRU over TE steps (masked) ----------------
    zero_f32<<<(B_ * H_ + 255) / 256, 256, 0, stream>>>(f_h, B_ * H_);
    for (int t = 0; t < TE_; ++t) {
        // gi = emb_w[enc[:,t]] @ w_ih_enc^T + b_ih_enc   (gather fused in GEMM)
        gemm(emb_w, E_, enc + t, TE_, w_ih_enc, E_, 1, b_ih_enc,
             f_gi, 3 * H_, B_, 3 * H_, E_);
        // gh = h @ w_hh_enc^T + b_hh_enc
        gemm(f_h, H_, nullptr, 0, w_hh_enc, H_, 1, b_hh_enc,
             f_gh, 3 * H_, B_, 3 * H_, H_);
        gru_combine<<<(B_ * H_ + 255) / 256, 256, 0, stream>>>(
            f_gi, f_gh, f_h, f_h, encLen, t, H_);
    }

    // origInfo / tranInfo = concat([sty, dom, h_enc])
    build_info<<<(B_ * HD_ + 255) / 256, 256, 0, stream>>>(
        label, tgtDom, sty_w, sty_b, dom_w, dom_b, f_h, f_or, f_tr);

    // ---------------- Teacher-forcing decoder ----------------
    for (int t = 0; t < T_; ++t) {
        const float* hp = t ? (f_hs + (size_t)(t - 1) * B_ * HD_) : f_or;
        gemm(emb_w, E_, dec + t, T_, w_ih_dec, E_, 1, b_ih_dec,
             f_gi, 3 * HD_, B_, 3 * HD_, E_);
        gemm(hp, HD_, nullptr, 0, w_hh_dec, HD_, 1, b_hh_dec,
             f_gh, 3 * HD_, B_, 3 * HD_, HD_);
        gru_combine<<<(B_ * HD_ + 255) / 256, 256, 0, stream>>>(
            f_gi, f_gh, hp, f_hs + (size_t)t * B_ * HD_, nullptr, t, HD_);
    }

    // ---------------- aeLoss: chunked logits + logsumexp NLL ----------------
    for (int c = 0; c < 8; ++c) {           // 8 timesteps (256 rows) per chunk
        gemm(f_hs + (size_t)c * 8 * B_ * HD_, HD_, nullptr, 0,
             proj_w, HD_, 1, proj_b, f_lg, V_, 8 * B_, V_, HD_);
        nll_rows<<<8 * B_, 256, 0, stream>>>(f_lg, c * 8, tar, decLen, f_nll);
    }
    ae_reduce<<<1, 256, 0, stream>>>(f_nll, outAe);

    // ---------------- Gumbel sampling decoder ----------------
    for (int t = 0; t < T_; ++t) {
        const float* hp = t ? f_hg : f_tr;
        if (t == 0)
            gemm(emb_w, E_, dec, T_, w_ih_dec, E_, 1, b_ih_dec,   // x0 = emb[dec[:,0]]
                 f_gi, 3 * HD_, B_, 3 * HD_, E_);
        else
            gemm(f_x, E_, nullptr, 0, w_ih_dec, E_, 1, b_ih_dec,
                 f_gi, 3 * HD_, B_, 3 * HD_, E_);
        gemm(hp, HD_, nullptr, 0, w_hh_dec, HD_, 1, b_hh_dec,
             f_gh, 3 * HD_, B_, 3 * HD_, HD_);
        gru_combine<<<(B_ * HD_ + 255) / 256, 256, 0, stream>>>(
            f_gi, f_gh, hp, f_hg, nullptr, t, HD_);

        // logits = h @ proj_w^T + proj_b
        gemm(f_hg, HD_, nullptr, 0, proj_w, HD_, 1, proj_b,
             f_lg, V_, B_, V_, HD_);
        gumbel_softmax<<<B_, 256, 0, stream>>>(
            f_lg, gumbel + (size_t)t * B_ * V_, f_pr, outTr, t);

        // x_next = proba @ emb_w   (W is K x N row-major here)
        gemm(f_pr, V_, nullptr, 0, emb_w, E_, 0, nullptr,
             f_x, E_, B_, E_, V_);
    }
}